// GraphTransformer_54065048323044
// MI455X (gfx1250) — compile-verified
//
#include <hip/hip_runtime.h>
#include <cstdint>
#include <cstddef>

typedef __attribute__((ext_vector_type(2))) float v2f;
typedef __attribute__((ext_vector_type(8))) float v8f;

// ---------- order-preserving float <-> uint encoding for atomicMax ----------
__device__ __forceinline__ unsigned enc_f32(float f) {
    unsigned u = __float_as_uint(f);
    return (u & 0x80000000u) ? ~u : (u | 0x80000000u);
}
__device__ __forceinline__ float dec_f32(unsigned u) {
    unsigned v = (u & 0x80000000u) ? (u & 0x7fffffffu) : ~u;
    return __uint_as_float(v);
}
__device__ __forceinline__ bool finite_f32(float f) {
    return (__float_as_uint(f) & 0x7f800000u) != 0x7f800000u;
}

template <int V> struct VecT;
template <> struct VecT<2> { typedef float2 T; };
template <> struct VecT<4> { typedef float4 T; };

// ---------------------------------------------------------------------------
// GEMM + bias:  out[N, M] = X[N, K] @ W[K, M] + b[M], fp32 exact via
// V_WMMA_F32_16X16X4_F32. One wave computes one 16x16 output tile.
// A (16x4 f32) lane layout: lanes 0-15 hold M=lane, K = kb+{0,1};
//                           lanes 16-31 hold M=lane-16, K = kb+{2,3}.
// B (4x16 f32) is the mirrored layout over columns.
// C/D: VGPR i, lanes 0-15 -> (M=i, N=lane); lanes 16-31 -> (M=i+8, N=lane-16).
// ---------------------------------------------------------------------------
__global__ void gemm_bias_wmma(const float* __restrict__ X,
                               const float* __restrict__ W,
                               const float* __restrict__ bias,
                               float* __restrict__ out,
                               int rowTiles, int K, int M) {
    const int lane   = threadIdx.x & 31;
    const int lm     = lane & 15;
    const int lhalf  = lane >> 4;
    const int colTiles = M >> 4;
    const int totalTiles = rowTiles * colTiles;
    int wid = blockIdx.x * (blockDim.x >> 5) + (threadIdx.x >> 5);
    const int nwaves = gridDim.x * (blockDim.x >> 5);

    for (int t = wid; t < totalTiles; t += nwaves) {
        const int tr = t / colTiles;
        const int tc = t - tr * colTiles;
        const int row = tr * 16 + lm;   // node row this lane feeds for A
        const int col = tc * 16 + lm;   // output column this lane feeds for B
        const float* xrow = X + (size_t)row * K;

        v8f acc = {};
        for (int kb = 0; kb < K; kb += 4) {
            const int ka = kb + 2 * lhalf;
            v2f a = *reinterpret_cast<const v2f*>(xrow + ka);  // K consecutive
            v2f b;
            b.x = W[(size_t)ka * M + col];
            b.y = W[(size_t)(ka + 1) * M + col];
            acc = __builtin_amdgcn_wmma_f32_16x16x4_f32(
                false, a, false, b, (short)0, acc, false, false);
        }
        const float bb = bias[col];
#pragma unroll
        for (int i = 0; i < 8; ++i) {
            const int r = tr * 16 + lhalf * 8 + i;
            out[(size_t)r * M + col] = acc[i] + bb;
        }
    }
}

// ---------------------------------------------------------------------------
// Edge pass 1: score[e,h] = dot(q[dst[e],h,:], k[src[e],h,:]) / sqrt(D)
// + segment max into mEnc (uint-encoded float atomicMax).
// One wave per edge; each lane owns HD/32 contiguous features.
// ---------------------------------------------------------------------------
template <int H, int D>
__global__ void edge_scores_kernel(const float* __restrict__ q,
                                   const float* __restrict__ k,
                                   const int* __restrict__ src,
                                   const int* __restrict__ dst,
                                   float* __restrict__ scores,
                                   unsigned* __restrict__ mEnc,
                                   int nedges) {
    constexpr int HD  = H * D;
    constexpr int VEC = HD / 32;
    constexpr int LPH = 32 / H;   // lanes per head
    typedef typename VecT<VEC>::T LT;
    const int lane = threadIdx.x & 31;
    int wid = blockIdx.x * (blockDim.x >> 5) + (threadIdx.x >> 5);
    const int nwaves = gridDim.x * (blockDim.x >> 5);
    const float scale = rsqrtf((float)D);

    for (int e = wid; e < nedges; e += nwaves) {
        const int s = src[e];
        const int d = dst[e];
        LT qv = *reinterpret_cast<const LT*>(q + (size_t)d * HD + lane * VEC);
        LT kv = *reinterpret_cast<const LT*>(k + (size_t)s * HD + lane * VEC);
        const float* qa = reinterpret_cast<const float*>(&qv);
        const float* ka = reinterpret_cast<const float*>(&kv);
        float acc = 0.f;
#pragma unroll
        for (int i = 0; i < VEC; ++i) acc += qa[i] * ka[i];
#pragma unroll
        for (int off = 1; off < LPH; off <<= 1) acc += __shfl_xor(acc, off, 32);
        const float sc = acc * scale;
        if ((lane & (LPH - 1)) == 0) {
            const int h = lane / LPH;
            scores[(size_t)e * H + h] = sc;
            atomicMax(&mEnc[(size_t)d * H + h], enc_f32(sc));
        }
    }
}

// ---------------------------------------------------------------------------
// Edge pass 2: p = exp(score - m[dst]);  ssum[dst,h] += p;
//              agg[dst,h,:] += p * v[src,h,:]   (unnormalized softmax-agg)
// ---------------------------------------------------------------------------
template <int H, int D>
__global__ void edge_accum_kernel(const float* __restrict__ v,
                                  const int* __restrict__ src,
                                  const int* __restrict__ dst,
                                  const float* __restrict__ scores,
                                  const unsigned* __restrict__ mEnc,
                                  float* __restrict__ ssum,
                                  float* __restrict__ agg,
                                  int nedges) {
    constexpr int HD  = H * D;
    constexpr int VEC = HD / 32;
    constexpr int LPH = 32 / H;
    typedef typename VecT<VEC>::T LT;
    const int lane = threadIdx.x & 31;
    int wid = blockIdx.x * (blockDim.x >> 5) + (threadIdx.x >> 5);
    const int nwaves = gridDim.x * (blockDim.x >> 5);

    for (int e = wid; e < nedges; e += nwaves) {
        const int s = src[e];
        const int d = dst[e];
        const int h = lane / LPH;
        const float sc = scores[(size_t)e * H + h];
        float m = dec_f32(mEnc[(size_t)d * H + h]);
        if (!finite_f32(m)) m = 0.f;           // empty-segment semantics
        const float p = __expf(sc - m);
        if ((lane & (LPH - 1)) == 0)
            atomicAdd(&ssum[(size_t)d * H + h], p);
        LT vv = *reinterpret_cast<const LT*>(v + (size_t)s * HD + lane * VEC);
        const float* va = reinterpret_cast<const float*>(&vv);
        float* ap = agg + (size_t)d * HD + lane * VEC;
#pragma unroll
        for (int i = 0; i < VEC; ++i) atomicAdd(ap + i, p * va[i]);
    }
}

// ---------------------------------------------------------------------------
// Node finalize: out = agg / (ssum + 1e-16) + skip   (+ optional ReLU)
// ---------------------------------------------------------------------------
template <int H, int D, bool RELU>
__global__ void finalize_kernel(const float* __restrict__ agg,
                                const float* __restrict__ ssum,
                                const float* __restrict__ skip,
                                float* __restrict__ out,
                                int nnodes) {
    constexpr int HD = H * D;
    long long idx = (long long)blockIdx.x * blockDim.x + threadIdx.x;
    const long long total = (long long)nnodes * HD;
    if (idx >= total) return;
    const int n = (int)(idx / HD);
    const int h = (int)((idx - (long long)n * HD) / D);
    float val = agg[idx] / (ssum[(size_t)n * H + h] + 1e-16f) + skip[idx];
    out[idx] = RELU ? fmaxf(val, 0.f) : val;
}

// ---------------------------------------------------------------------------
extern "C" void kernel_launch(void* const* d_in, const int* in_sizes, int n_in,
                              void* d_out, int out_size, void* d_ws, size_t ws_size,
                              hipStream_t stream) {
    (void)n_in; (void)out_size; (void)ws_size;
    const float* x   = (const float*)d_in[0];
    const float* Wq1 = (const float*)d_in[1];  const float* bq1 = (const float*)d_in[2];
    const float* Wk1 = (const float*)d_in[3];  const float* bk1 = (const float*)d_in[4];
    const float* Wv1 = (const float*)d_in[5];  const float* bv1 = (const float*)d_in[6];
    const float* Ws1 = (const float*)d_in[7];  const float* bs1 = (const float*)d_in[8];
    const float* Wq2 = (const float*)d_in[9];  const float* bq2 = (const float*)d_in[10];
    const float* Wk2 = (const float*)d_in[11]; const float* bk2 = (const float*)d_in[12];
    const float* Wv2 = (const float*)d_in[13]; const float* bv2 = (const float*)d_in[14];
    const float* Ws2 = (const float*)d_in[15]; const float* bs2 = (const float*)d_in[16];
    const int* esrc  = (const int*)d_in[17];
    const int* edst  = (const int*)d_in[18];

    const int N = in_sizes[0] / 128;   // 50000
    const int E = in_sizes[17];        // 800000
    const size_t NN = (size_t)N;

    // ---- workspace layout (floats) ----
    float* ws = (float*)d_ws;
    float* q1 = ws;                                 // [N,128]
    float* k1 = q1 + NN * 128;                      // [N,128]
    float* v1 = k1 + NN * 128;                      // [N,128]
    float* s1 = v1 + NN * 128;                      // [N,128] skip proj
    float* scores = s1 + NN * 128;                  // [E,8]
    unsigned* mEnc = (unsigned*)(scores + (size_t)E * 8);  // [N,8]
    float* ssum = (float*)mEnc + NN * 8;            // [N,8]
    float* agg  = ssum + NN * 8;                    // [N,128]
    float* hbuf = agg + NN * 128;                   // [N,128] layer-1 output
    const size_t accumBytes = (size_t)(8 + 8 + 128) * NN * sizeof(float);

    const dim3 blk(256);
    const int rowTiles = N / 16;                    // 3125 (exact)
    const int edgeBlocks = (E + 7) / 8;             // one wave per edge

    // ---- layer 1: projections (M=128) ----
    const int gB1 = (rowTiles * 8 + 7) / 8;
    gemm_bias_wmma<<<gB1, blk, 0, stream>>>(x, Wq1, bq1, q1, rowTiles, 128, 128);
    gemm_bias_wmma<<<gB1, blk, 0, stream>>>(x, Wk1, bk1, k1, rowTiles, 128, 128);
    gemm_bias_wmma<<<gB1, blk, 0, stream>>>(x, Wv1, bv1, v1, rowTiles, 128, 128);
    gemm_bias_wmma<<<gB1, blk, 0, stream>>>(x, Ws1, bs1, s1, rowTiles, 128, 128);

    hipMemsetAsync(mEnc, 0, accumBytes, stream);    // mEnc=0 (decodes NaN), ssum=0, agg=0

    edge_scores_kernel<8, 16><<<edgeBlocks, blk, 0, stream>>>(q1, k1, esrc, edst, scores, mEnc, E);
    edge_accum_kernel<8, 16><<<edgeBlocks, blk, 0, stream>>>(v1, esrc, edst, scores, mEnc, ssum, agg, E);
    {
        const long long tot = (long long)N * 128;
        finalize_kernel<8, 16, true><<<(unsigned)((tot + 255) / 256), blk, 0, stream>>>(
            agg, ssum, s1, hbuf, N);
    }

    // ---- layer 2: projections (M=64), reuse proj region ----
    float* q2 = ws;
    float* k2 = q2 + NN * 64;
    float* v2 = k2 + NN * 64;
    float* s2 = v2 + NN * 64;
    const int gB2 = (rowTiles * 4 + 7) / 8;
    gemm_bias_wmma<<<gB2, blk, 0, stream>>>(hbuf, Wq2, bq2, q2, rowTiles, 128, 64);
    gemm_bias_wmma<<<gB2, blk, 0, stream>>>(hbuf, Wk2, bk2, k2, rowTiles, 128, 64);
    gemm_bias_wmma<<<gB2, blk, 0, stream>>>(hbuf, Wv2, bv2, v2, rowTiles, 128, 64);
    gemm_bias_wmma<<<gB2, blk, 0, stream>>>(hbuf, Ws2, bs2, s2, rowTiles, 128, 64);

    hipMemsetAsync(mEnc, 0, accumBytes, stream);

    edge_scores_kernel<1, 64><<<edgeBlocks, blk, 0, stream>>>(q2, k2, esrc, edst, scores, mEnc, E);
    edge_accum_kernel<1, 64><<<edgeBlocks, blk, 0, stream>>>(v2, esrc, edst, scores, mEnc, ssum, agg, E);
    {
        const long long tot = (long long)N * 64;
        finalize_kernel<1, 64, false><<<(unsigned)((tot + 255) / 256), blk, 0, stream>>>(
            agg, ssum, s2, (float*)d_out, N);
    }
}